// Model_36636071034893
// MI455X (gfx1250) — compile-verified
//
#include <hip/hip_runtime.h>
#include <hip/hip_bf16.h>

#define FEAT   128
#define LDSP   129   // +1 pad: LDS has 64 banks, row stride 129 decorrelates
#define BN_EPS 1e-5f
#define ROWS_PER_BN_BLOCK 256

typedef __attribute__((ext_vector_type(2))) float v2f;
typedef __attribute__((ext_vector_type(8))) float v8f;

// ---------------------------------------------------------------- utilities
__global__ void k_zero(float* __restrict__ p, long long n) {
    long long i = (long long)blockIdx.x * blockDim.x + threadIdx.x;
    long long stride = (long long)gridDim.x * blockDim.x;
    for (; i < n; i += stride) p[i] = 0.0f;
}

__global__ void k_fill1(float* __restrict__ p, long long n) {
    long long i = (long long)blockIdx.x * blockDim.x + threadIdx.x;
    long long stride = (long long)gridDim.x * blockDim.x;
    for (; i < n; i += stride) p[i] = 1.0f;   // self-loop contributes 1 to degree
}

// degree accumulation over target index (col)
__global__ void k_deg_accum(const long long* __restrict__ cols, float* __restrict__ deg,
                            long long E) {
    long long i = (long long)blockIdx.x * blockDim.x + threadIdx.x;
    long long stride = (long long)gridDim.x * blockDim.x;
    for (; i < E; i += stride) atomicAdd(&deg[cols[i]], 1.0f);
}

__global__ void k_dinv(const float* __restrict__ deg, float* __restrict__ dinv, long long n) {
    long long i = (long long)blockIdx.x * blockDim.x + threadIdx.x;
    long long stride = (long long)gridDim.x * blockDim.x;
    for (; i < n; i += stride) dinv[i] = rsqrtf(deg[i]);
}

// ---------------------------------------------------------------- WMMA GEMM
// out[N,128] = h[N,128] @ W[128,128], fp32 via V_WMMA_F32_16X16X4_F32.
// One workgroup = 16 output rows; 8 waves each own a 16-wide N tile.
// K is walked in 8 panels of 16, staged through LDS.
__global__ __launch_bounds__(256) void k_gemm(const float* __restrict__ h,
                                              const float* __restrict__ W,
                                              float* __restrict__ out, int n) {
    __shared__ float As[16][LDSP];
    __shared__ float Bs[16][LDSP];

    const int tid  = threadIdx.x;
    const int row0 = blockIdx.x * 16;

    // stage the 16x128 A strip (2048 floats / 256 threads = 2x float4 each)
    for (int idx = tid; idx < 16 * FEAT / 4; idx += 256) {
        int r = (idx * 4) / FEAT, c = (idx * 4) % FEAT;
        int gr = row0 + r;
        float4 v = {0.f, 0.f, 0.f, 0.f};
        if (gr < n) v = ((const float4*)(h + (size_t)gr * FEAT))[c >> 2];
        As[r][c + 0] = v.x; As[r][c + 1] = v.y; As[r][c + 2] = v.z; As[r][c + 3] = v.w;
    }

    const int wave = tid >> 5;
    const int lane = tid & 31;
    const int half = lane >> 4;   // A: VGPR pair holds K pair {0,1} (lanes<16) or {2,3}
    const int l16  = lane & 15;
    const int nb   = wave * 16;   // this wave's N tile base

    v8f acc = {};

    for (int p = 0; p < FEAT / 16; ++p) {          // 8 K-panels of 16
        __syncthreads();
        // stage W[p*16 .. p*16+15][0..127] (2048 floats)
        for (int idx = tid; idx < 16 * FEAT / 4; idx += 256) {
            int r = (idx * 4) / FEAT, c = (idx * 4) % FEAT;
            float4 v = ((const float4*)(W + (size_t)(p * 16 + r) * FEAT))[c >> 2];
            Bs[r][c + 0] = v.x; Bs[r][c + 1] = v.y; Bs[r][c + 2] = v.z; Bs[r][c + 3] = v.w;
        }
        __syncthreads();

#pragma unroll
        for (int k0 = 0; k0 < 16; k0 += 4) {
            const int ko = k0 + 2 * half;          // K offset within panel for this half
            v2f a, b;
            a.x = As[l16][p * 16 + ko];            // A: row M=l16, K=ko
            a.y = As[l16][p * 16 + ko + 1];        //            , K=ko+1
            b.x = Bs[ko][nb + l16];                // B: K=ko,   N=nb+l16 (col per lane)
            b.y = Bs[ko + 1][nb + l16];            //    K=ko+1
            acc = __builtin_amdgcn_wmma_f32_16x16x4_f32(
                false, a, false, b, (short)0, acc, false, false);
        }
    }

    // C/D layout: VGPR r -> M=r (lanes 0-15), M=r+8 (lanes 16-31); N=l16
#pragma unroll
    for (int r = 0; r < 8; ++r) {
        int gr = row0 + r + 8 * half;
        if (gr < n) out[(size_t)gr * FEAT + nb + l16] = acc[r];
    }
}

// ---------------------------------------------------------------- GCN scatter
// conv[i][f] = bias[f] + xw[i][f] * dinv[i]^2   (self-loop + bias folded in)
__global__ void k_conv_init(const float* __restrict__ xw, const float* __restrict__ dinv,
                            const float* __restrict__ bias, float* __restrict__ conv,
                            long long total) {
    long long i = (long long)blockIdx.x * blockDim.x + threadIdx.x;
    long long stride = (long long)gridDim.x * blockDim.x;
    for (; i < total; i += stride) {
        int f = (int)(i & (FEAT - 1));
        long long node = i >> 7;
        float di = dinv[node];
        conv[i] = bias[f] + xw[i] * di * di;
    }
}

// one wave per edge: 32 lanes x float4 = 128 features, float atomics into conv[dst]
__global__ __launch_bounds__(256) void k_scatter(const float* __restrict__ xw,
                                                 float* __restrict__ conv,
                                                 const long long* __restrict__ rows,
                                                 const long long* __restrict__ cols,
                                                 const float* __restrict__ dinv,
                                                 long long E) {
    long long wid = ((long long)blockIdx.x * blockDim.x + threadIdx.x) >> 5;
    long long nw  = ((long long)gridDim.x * blockDim.x) >> 5;
    const int lane = threadIdx.x & 31;
    for (long long e = wid; e < E; e += nw) {
        long long s = rows[e], d = cols[e];
        float w = dinv[s] * dinv[d];
        float4 v = ((const float4*)(xw + (size_t)s * FEAT))[lane];
        float* dst = conv + (size_t)d * FEAT + lane * 4;
        atomicAdd(dst + 0, v.x * w);
        atomicAdd(dst + 1, v.y * w);
        atomicAdd(dst + 2, v.z * w);
        atomicAdd(dst + 3, v.w * w);
    }
}

// ---------------------------------------------------------------- BatchNorm
// per-feature partial sums of z = prev + conv; 128 threads = 1 thread/feature
__global__ __launch_bounds__(128) void k_bn_stats(const float* __restrict__ conv,
                                                  const float* __restrict__ prev,
                                                  float* __restrict__ stats, int n) {
    const int f = threadIdx.x;
    int r0 = blockIdx.x * ROWS_PER_BN_BLOCK;
    int r1 = r0 + ROWS_PER_BN_BLOCK; if (r1 > n) r1 = n;
    const bool hasprev = (prev != nullptr);
    float s = 0.f, sq = 0.f;
    for (int r = r0; r < r1; ++r) {
        float z = conv[(size_t)r * FEAT + f];
        if (hasprev) z += prev[(size_t)r * FEAT + f];
        s += z; sq += z * z;
    }
    atomicAdd(&stats[f], s);
    atomicAdd(&stats[FEAT + f], sq);
}

__global__ __launch_bounds__(128) void k_bn_final(const float* __restrict__ stats,
                                                  float* __restrict__ bnp, float n) {
    int f = threadIdx.x;
    float mu  = stats[f] / n;
    float var = stats[FEAT + f] / n - mu * mu;
    bnp[f] = mu;
    bnp[FEAT + f] = rsqrtf(var + BN_EPS);
}

// h_out = elu((z - mu) * rstd * gamma + beta)
__global__ void k_bn_apply(const float* __restrict__ conv, const float* __restrict__ prev,
                           const float* __restrict__ bnp, const float* __restrict__ gamma,
                           const float* __restrict__ beta, float* __restrict__ hout,
                           long long total) {
    long long i = (long long)blockIdx.x * blockDim.x + threadIdx.x;
    long long stride = (long long)gridDim.x * blockDim.x;
    const bool hasprev = (prev != nullptr);
    for (; i < total; i += stride) {
        int f = (int)(i & (FEAT - 1));
        float z = conv[i];
        if (hasprev) z += prev[i];
        float zn = (z - bnp[f]) * bnp[FEAT + f] * gamma[f] + beta[f];
        hout[i] = zn > 0.f ? zn : expm1f(zn);
    }
}

// ---------------------------------------------------------------- pooling
__global__ __launch_bounds__(256) void k_pool_accum(const float* __restrict__ h,
                                                    const long long* __restrict__ batch,
                                                    float* __restrict__ pooled,
                                                    float* __restrict__ counts, long long n) {
    long long wid = ((long long)blockIdx.x * blockDim.x + threadIdx.x) >> 5;
    long long nw  = ((long long)gridDim.x * blockDim.x) >> 5;
    const int lane = threadIdx.x & 31;
    for (long long node = wid; node < n; node += nw) {
        long long g = batch[node];
        float4 v = ((const float4*)(h + (size_t)node * FEAT))[lane];
        float* dst = pooled + (size_t)g * FEAT + lane * 4;
        atomicAdd(dst + 0, v.x);
        atomicAdd(dst + 1, v.y);
        atomicAdd(dst + 2, v.z);
        atomicAdd(dst + 3, v.w);
        if (lane == 0) atomicAdd(&counts[g], 1.0f);
    }
}

__global__ void k_readout(const float* __restrict__ pooled, const float* __restrict__ counts,
                          const float* __restrict__ Wro, const float* __restrict__ bro,
                          float* __restrict__ out, int ngraphs) {
    int t = blockIdx.x * blockDim.x + threadIdx.x;
    if (t >= ngraphs * 2) return;
    int g = t >> 1, o = t & 1;
    float c = fmaxf(counts[g], 1.0f);
    float acc = bro[o];
    for (int f = 0; f < FEAT; ++f)
        acc += (pooled[(size_t)g * FEAT + f] / c) * Wro[f * 2 + o];
    out[t] = acc;
}

// ---------------------------------------------------------------- host
extern "C" void kernel_launch(void* const* d_in, const int* in_sizes, int n_in,
                              void* d_out, int out_size, void* d_ws, size_t ws_size,
                              hipStream_t stream) {
    const float*     x      = (const float*)d_in[0];
    const long long* ei     = (const long long*)d_in[1];   // [2, E] int64
    const long long* batch  = (const long long*)d_in[2];   // [N] int64
    const float*     Ws     = (const float*)d_in[3];       // [3,128,128]
    const float*     bs     = (const float*)d_in[4];       // [3,128]
    const float*     gammas = (const float*)d_in[5];
    const float*     betas  = (const float*)d_in[6];
    const float*     Wro    = (const float*)d_in[7];       // [128,2]
    const float*     bro    = (const float*)d_in[8];
    float*           out    = (float*)d_out;

    const int       N  = in_sizes[0] / FEAT;
    const long long E  = in_sizes[1] / 2;
    const int       NB = in_sizes[3] / (FEAT * FEAT);
    const int       NG = out_size / 2;

    const long long* rows = ei;          // source
    const long long* cols = ei + E;      // target (aggregation index)

    // workspace carve-out
    float* ws = (float*)d_ws;
    const size_t NF = (size_t)N * FEAT;
    float* B0     = ws;            ws += NF;
    float* B1     = ws;            ws += NF;
    float* B2     = ws;            ws += NF;
    float* conv   = ws;            ws += NF;
    float* deg    = ws;            ws += N;
    float* dinv   = ws;            ws += N;
    float* stats  = ws;            ws += 2 * FEAT;
    float* bnp    = ws;            ws += 2 * FEAT;
    float* pooled = ws;            ws += (size_t)NG * FEAT;
    float* counts = ws;            ws += NG;

    const long long total = (long long)NF;
    const int TB = 256;
    const int gElem  = (int)((total + TB - 1) / TB);
    const int gNode  = (N + TB - 1) / TB;
    const int gEdgeT = (int)((E + TB - 1) / TB);
    const int gEdgeW = (int)((E * 32 + TB - 1) / TB);   // wave per edge
    const int gNodeW = (int)(((long long)N * 32 + TB - 1) / TB);
    const int gGemm  = (N + 15) / 16;
    const int gBN    = (N + ROWS_PER_BN_BLOCK - 1) / ROWS_PER_BN_BLOCK;

    // degrees (init to 1 for self-loops) -> dinv
    k_fill1<<<gNode, TB, 0, stream>>>(deg, N);
    k_deg_accum<<<gEdgeT, TB, 0, stream>>>(cols, deg, E);
    k_dinv<<<gNode, TB, 0, stream>>>(deg, dinv, N);

    // buffer rotation implementing the prev/h ping-pong:
    //   i=0: gemm(x)->B0,  apply(prev=0)  -> B1
    //   i=1: gemm(B1)->B0, apply(prev=x)  -> B2
    //   i=2: gemm(B2)->B0, apply(prev=B1) -> B0
    const float* h_in[3]  = { x,      B1, B2 };
    const float* prev[3]  = { nullptr, x, B1 };
    float*       h_out[3] = { B1,     B2, B0 };

    for (int i = 0; i < NB && i < 3; ++i) {
        const float* W  = Ws     + (size_t)i * FEAT * FEAT;
        const float* bi = bs     + (size_t)i * FEAT;
        const float* gi = gammas + (size_t)i * FEAT;
        const float* bt = betas  + (size_t)i * FEAT;

        k_gemm<<<gGemm, 256, 0, stream>>>(h_in[i], W, B0, N);
        k_conv_init<<<gElem, TB, 0, stream>>>(B0, dinv, bi, conv, total);
        k_scatter<<<gEdgeW, TB, 0, stream>>>(B0, conv, rows, cols, dinv, E);

        k_zero<<<1, 2 * FEAT, 0, stream>>>(stats, 2 * FEAT);
        k_bn_stats<<<gBN, FEAT, 0, stream>>>(conv, prev[i], stats, N);
        k_bn_final<<<1, FEAT, 0, stream>>>(stats, bnp, (float)N);
        k_bn_apply<<<gElem, TB, 0, stream>>>(conv, prev[i], bnp, gi, bt, h_out[i], total);
    }

    // global mean pool + readout
    k_zero<<<(NG * FEAT + TB - 1) / TB, TB, 0, stream>>>(pooled, (long long)NG * FEAT);
    k_zero<<<1, NG, 0, stream>>>(counts, NG);
    k_pool_accum<<<gNodeW, TB, 0, stream>>>(B0, batch, pooled, counts, N);
    k_readout<<<1, 2 * NG, 0, stream>>>(pooled, counts, Wro, bro, out, NG);
}